// GCN_13374528160099
// MI455X (gfx1250) — compile-verified
//
#include <hip/hip_runtime.h>

typedef __attribute__((ext_vector_type(2))) float v2f;
typedef __attribute__((ext_vector_type(8))) float v8f;

#define N_NODES 50000
#define E_EDGES 800000
#define F_IN    512
#define HID     128
#define OUT_F   40

// ---------------------------------------------------------------------------
// Init: H = broadcast(b1), out = broadcast(b2). SpMM atomics then accumulate
// on top of the bias, fusing "+ b" for free.
// ---------------------------------------------------------------------------
__global__ void k_init_bias(float* __restrict__ H, const float* __restrict__ b1,
                            float* __restrict__ out, const float* __restrict__ b2) {
  const long long nh = (long long)N_NODES * HID;
  const long long no = (long long)N_NODES * OUT_F;
  long long i = (long long)blockIdx.x * blockDim.x + threadIdx.x;
  if (i < nh) {
    H[i] = b1[i & (HID - 1)];
  } else if (i < nh + no) {
    long long j = i - nh;
    out[j] = b2[(int)(j % OUT_F)];
  }
}

// ---------------------------------------------------------------------------
// GEMM1: H0[50000,128] = X[50000,512] @ W1[512,128], fp32 WMMA 16x16x4.
// Grid: 3125 blocks x 256 threads. Wave w of a block owns column tile w (8
// tiles = 128 cols); all 8 waves share the same 16 rows of X (cache reuse).
// ---------------------------------------------------------------------------
__global__ void k_gemm1(const float* __restrict__ X, const float* __restrict__ W1,
                        float* __restrict__ H0) {
  const int wave = threadIdx.x >> 5;      // 0..7 -> column tile
  const int lane = threadIdx.x & 31;
  const int half = lane >> 4;             // K sub-pair select
  const int r    = lane & 15;
  const int rowTile = blockIdx.x;         // 0..3124 (50000/16 exact)
  const int row = rowTile * 16 + r;       // A: M = lane%16
  const int col = wave * 16 + r;          // B/D: N = lane%16

  const float* __restrict__ xrow = X + (long long)row * F_IN;
  v8f acc = {};

#pragma unroll 4
  for (int k = 0; k < F_IN; k += 4) {
    const int ka = k + 2 * half;
    v2f a, b;
    a.x = xrow[ka];
    a.y = xrow[ka + 1];
    b.x = W1[ka * HID + col];
    b.y = W1[(ka + 1) * HID + col];
    acc = __builtin_amdgcn_wmma_f32_16x16x4_f32(
        /*neg_a=*/false, a, /*neg_b=*/false, b,
        /*c_mod=*/(short)0, acc, /*reuse_a=*/false, /*reuse_b=*/false);
  }

  // D layout: VGPR v holds row (v + 8*half), col = lane%16
  float* __restrict__ hb = H0 + (long long)(rowTile * 16 + 8 * half) * HID + wave * 16 + r;
#pragma unroll
  for (int v = 0; v < 8; ++v) hb[v * HID] = acc[v];
}

// ---------------------------------------------------------------------------
// SpMM1: H[dst] += w_e * H0[src]  (128 f32 per edge; one wave32 per edge,
// float4 per lane -> coalesced 512B gather, atomics land in L2 since H fits).
// ---------------------------------------------------------------------------
__global__ void k_spmm1(const int* __restrict__ esrc, const int* __restrict__ edst,
                        const float* __restrict__ ew,
                        const float* __restrict__ H0, float* __restrict__ H) {
  const int e = blockIdx.x * 8 + (threadIdx.x >> 5);
  if (e >= E_EDGES) return;               // wave-uniform
  const int lane = threadIdx.x & 31;
  const int s = esrc[e];
  const int d = edst[e];
  const float we = ew[e];
  const float4 v = ((const float4*)(H0 + (long long)s * HID))[lane];
  float* __restrict__ hd = H + (long long)d * HID + lane * 4;
  atomicAdd(hd + 0, we * v.x);
  atomicAdd(hd + 1, we * v.y);
  atomicAdd(hd + 2, we * v.z);
  atomicAdd(hd + 3, we * v.w);
}

// ---------------------------------------------------------------------------
// GEMM2 fused: G[50000,40] = (relu(H) * drop_mask) @ W2[128,40].
// ReLU + dropout applied on the A-operand load (no extra pass over 25MB).
// Output padded to 3 column tiles of 16 (48); lane guard on B load / D store.
// ---------------------------------------------------------------------------
__global__ void k_gemm2(const float* __restrict__ H, const float* __restrict__ mask,
                        const float* __restrict__ W2, float* __restrict__ G) {
  const int id = blockIdx.x * 8 + (threadIdx.x >> 5);   // tile id, wave-uniform
  if (id >= 3125 * 3) return;
  const int lane = threadIdx.x & 31;
  const int half = lane >> 4;
  const int r    = lane & 15;
  const int rowTile = id / 3;
  const int colTile = id % 3;
  const int row = rowTile * 16 + r;
  const int col = colTile * 16 + r;
  const bool colOK = (col < OUT_F);

  const float* __restrict__ hrow = H + (long long)row * HID;
  const float* __restrict__ mrow = mask + (long long)row * HID;
  v8f acc = {};

#pragma unroll 4
  for (int k = 0; k < HID; k += 4) {
    const int ka = k + 2 * half;
    float h0 = hrow[ka], h1 = hrow[ka + 1];
    v2f a, b;
    a.x = (h0 > 0.0f ? h0 : 0.0f) * mrow[ka];
    a.y = (h1 > 0.0f ? h1 : 0.0f) * mrow[ka + 1];
    b.x = colOK ? W2[ka * OUT_F + col] : 0.0f;
    b.y = colOK ? W2[(ka + 1) * OUT_F + col] : 0.0f;
    acc = __builtin_amdgcn_wmma_f32_16x16x4_f32(
        false, a, false, b, (short)0, acc, false, false);
  }

  if (colOK) {
    float* __restrict__ g = G + (long long)(rowTile * 16 + 8 * half) * OUT_F + col;
#pragma unroll
    for (int v = 0; v < 8; ++v) g[v * OUT_F] = acc[v];
  }
}

// ---------------------------------------------------------------------------
// SpMM2: out[dst] += w_e * G[src]  (40 f32 per edge; wave32 per edge,
// lanes 0..31 cover cols 0..31, lanes 0..7 additionally cover 32..39).
// ---------------------------------------------------------------------------
__global__ void k_spmm2(const int* __restrict__ esrc, const int* __restrict__ edst,
                        const float* __restrict__ ew,
                        const float* __restrict__ G, float* __restrict__ out) {
  const int e = blockIdx.x * 8 + (threadIdx.x >> 5);
  if (e >= E_EDGES) return;               // wave-uniform
  const int lane = threadIdx.x & 31;
  const int s = esrc[e];
  const int d = edst[e];
  const float we = ew[e];
  const float* __restrict__ g = G + (long long)s * OUT_F;
  float* __restrict__ o = out + (long long)d * OUT_F;
  atomicAdd(o + lane, we * g[lane]);
  if (lane < OUT_F - 32) atomicAdd(o + lane + 32, we * g[lane + 32]);
}

// ---------------------------------------------------------------------------
extern "C" void kernel_launch(void* const* d_in, const int* in_sizes, int n_in,
                              void* d_out, int out_size, void* d_ws, size_t ws_size,
                              hipStream_t stream) {
  const float* x    = (const float*)d_in[0];   // [50000,512]
  const float* W1   = (const float*)d_in[1];   // [512,128]
  const float* b1   = (const float*)d_in[2];   // [128]
  const float* W2   = (const float*)d_in[3];   // [128,40]
  const float* b2   = (const float*)d_in[4];   // [40]
  const float* ew   = (const float*)d_in[5];   // [800000]
  const float* mask = (const float*)d_in[6];   // [50000,128]
  const int* esrc   = (const int*)d_in[7];     // [800000]
  const int* edst   = (const int*)d_in[8];     // [800000]
  float* out = (float*)d_out;                  // [50000,40]

  // Workspace carve-up: H0 | H | G
  float* H0 = (float*)d_ws;                              // 50000*128
  float* H  = H0 + (long long)N_NODES * HID;             // 50000*128
  float* G  = H  + (long long)N_NODES * HID;             // 50000*40

  // 1) bias pre-init of accumulation targets
  {
    long long total = (long long)N_NODES * HID + (long long)N_NODES * OUT_F;
    int blocks = (int)((total + 255) / 256);
    k_init_bias<<<blocks, 256, 0, stream>>>(H, b1, out, b2);
  }
  // 2) H0 = X @ W1
  k_gemm1<<<N_NODES / 16, 256, 0, stream>>>(x, W1, H0);
  // 3) H += adj-scatter of H0 (lands on b1)
  k_spmm1<<<E_EDGES / 8, 256, 0, stream>>>(esrc, edst, ew, H0, H);
  // 4) G = (relu(H)*mask) @ W2
  k_gemm2<<<(3125 * 3 + 7) / 8, 256, 0, stream>>>(H, mask, W2, G);
  // 5) out += adj-scatter of G (lands on b2)
  k_spmm2<<<E_EDGES / 8, 256, 0, stream>>>(esrc, edst, ew, G, out);
}